// SeparationLoss_62362925138704
// MI455X (gfx1250) — compile-verified
//
#include <hip/hip_runtime.h>

typedef __attribute__((ext_vector_type(2))) float v2f;
typedef __attribute__((ext_vector_type(8))) float v8f;

#define NBLOCKS 2048
#define BLOCK   256
#define WPB     (BLOCK / 32)   // waves per block (wave32)

// Stage 1: per-batch separation loss via one V_WMMA_F32_16X16X4_F32 Gram per
// batch (joints 0..15) + per-lane handling of joint 16; deterministic
// wave/block reduction into one partial per block.
__global__ __launch_bounds__(BLOCK) void sep_loss_partial(
    const float* __restrict__ kps, float* __restrict__ partial, int nb) {
  const int lane   = threadIdx.x & 31;
  const int wib    = threadIdx.x >> 5;
  const int wave   = blockIdx.x * WPB + wib;
  const int nwaves = gridDim.x * WPB;
  const int half   = lane >> 4;      // 0: K=0,1 slice; 1: K=2,3 slice
  const int n      = lane & 15;      // column / own-joint index

  float acc = 0.0f;

  for (int b = wave; b < nb; b += nwaves) {
    const float* base = kps + (size_t)b * 51;  // 17 joints * 3 coords

    // Prefetch this wave's next batch (speculative, RT).
    __builtin_prefetch(base + (size_t)nwaves * 51, 0, 0);

    // Joint n coords (lanes 0-15 and 16-31 load the same joint -> broadcast).
    float kx = base[n * 3 + 0];
    float ky = base[n * 3 + 1];
    float kz = base[n * 3 + 2];
    // Joint 16 coords (wave-uniform address -> cache broadcast).
    float px = base[48], py = base[49], pz = base[50];

    float sq_n = kx * kx + ky * ky + kz * kz;

    // A == B fragment for X * X^T (16x4 f32, K padded with 0):
    //   lanes 0-15 : v0 = x (K=0), v1 = y (K=1)
    //   lanes 16-31: v0 = z (K=2), v1 = 0 (K=3)
    v2f frag;
    frag.x = half ? kz : kx;
    frag.y = half ? 0.0f : ky;

    v8f c = {};
    v8f g = __builtin_amdgcn_wmma_f32_16x16x4_f32(
        /*neg_a=*/false, frag, /*neg_b=*/false, frag,
        /*c_mod=*/(short)0, c, /*reuse_a=*/false, /*reuse_b=*/false);

    // Lane holds gram[m][n] for m = r + 8*half, r = 0..7.
#pragma unroll
    for (int r = 0; r < 8; ++r) {
      int m = r + (half << 3);
      float sq_m = __shfl(sq_n, m, 32);          // sq of row joint m
      float d2 = sq_m + sq_n - 2.0f * g[r];
      float h = fmaxf(0.0f, 1.0f - d2);          // threshold^2 = 1
      acc += (m != n) ? h : 0.0f;                // drop diagonal
    }

    // Joint-16 vs joint-n: both lane halves compute it -> ordered-pair x2.
    float dx = kx - px, dy = ky - py, dz = kz - pz;
    float dd = dx * dx + dy * dy + dz * dz;
    acc += fmaxf(0.0f, 1.0f - dd);
  }

  // Deterministic wave32 butterfly reduction.
#pragma unroll
  for (int off = 16; off >= 1; off >>= 1)
    acc += __shfl_xor(acc, off, 32);

  __shared__ float sdata[WPB];
  if (lane == 0) sdata[wib] = acc;
  __syncthreads();
  if (threadIdx.x == 0) {
    float s = 0.0f;
#pragma unroll
    for (int w = 0; w < WPB; ++w) s += sdata[w];  // fixed order
    partial[blockIdx.x] = s;
  }
}

// Stage 2: single-block deterministic tree reduction of block partials.
__global__ __launch_bounds__(256) void sep_loss_finish(
    const float* __restrict__ partial, float* __restrict__ out, int nparts,
    float inv_nb) {
  __shared__ float sdata[256];
  float s = 0.0f;
  for (int i = threadIdx.x; i < nparts; i += 256) s += partial[i];  // fixed order
  sdata[threadIdx.x] = s;
  __syncthreads();
#pragma unroll
  for (int off = 128; off >= 1; off >>= 1) {
    if (threadIdx.x < off) sdata[threadIdx.x] += sdata[threadIdx.x + off];
    __syncthreads();
  }
  if (threadIdx.x == 0) out[0] = sdata[0] * inv_nb;
}

extern "C" void kernel_launch(void* const* d_in, const int* in_sizes, int n_in,
                              void* d_out, int out_size, void* d_ws,
                              size_t ws_size, hipStream_t stream) {
  (void)n_in; (void)out_size; (void)ws_size;
  const float* kps = (const float*)d_in[0];
  float* out = (float*)d_out;
  float* partial = (float*)d_ws;  // NBLOCKS floats of scratch

  const int nb = in_sizes[0] / 51;  // batches (17 joints * 3 coords)

  sep_loss_partial<<<NBLOCKS, BLOCK, 0, stream>>>(kps, partial, nb);
  sep_loss_finish<<<1, 256, 0, stream>>>(partial, out, NBLOCKS,
                                         1.0f / (float)nb);
}